// GNNBlock_25709674233976
// MI455X (gfx1250) — compile-verified
//
#include <hip/hip_runtime.h>

typedef __attribute__((ext_vector_type(2))) float v2f;
typedef __attribute__((ext_vector_type(8))) float v8f;

#define D_FEAT 128

// ---------------------------------------------------------------------------
// Kernel 1: h = node_feat (fresh accumulator every call; graph-replay safe)
// ---------------------------------------------------------------------------
__global__ void init_h_kernel(const float* __restrict__ node_feat,
                              float* __restrict__ h, int n4) {
    int i = blockIdx.x * blockDim.x + threadIdx.x;
    if (i < n4) {
        ((float4*)h)[i] = ((const float4*)node_feat)[i];
    }
}

// ---------------------------------------------------------------------------
// Kernel 2: for each edge: h[dst] += relu(node_feat[src] + edge_attr)
// One wave (32 lanes) per edge, 4 floats per lane (D=128).
// unsafeAtomicAdd -> hardware global_atomic_add_f32 (no CAS loop); the
// 50000x128 fp32 target lives in the 192MB L2.
// ---------------------------------------------------------------------------
__global__ void edge_msg_kernel(const float* __restrict__ node_feat,
                                const int* __restrict__ src,
                                const int* __restrict__ dst,
                                const float* __restrict__ edge_attr,
                                float* __restrict__ h, int nEdges) {
    int e = blockIdx.x * (blockDim.x >> 5) + (threadIdx.x >> 5);
    if (e >= nEdges) return;
    int lane = threadIdx.x & 31;

    size_t s = (size_t)src[e];
    size_t d = (size_t)dst[e];

    float4 x  = *(const float4*)(node_feat + s * D_FEAT + lane * 4);
    float4 ea = *(const float4*)(edge_attr + (size_t)e * D_FEAT + lane * 4);

    float4 m;
    m.x = fmaxf(x.x + ea.x, 0.0f);
    m.y = fmaxf(x.y + ea.y, 0.0f);
    m.z = fmaxf(x.z + ea.z, 0.0f);
    m.w = fmaxf(x.w + ea.w, 0.0f);

    float* out = h + d * D_FEAT + lane * 4;
    unsafeAtomicAdd(out + 0, m.x);
    unsafeAtomicAdd(out + 1, m.y);
    unsafeAtomicAdd(out + 2, m.z);
    unsafeAtomicAdd(out + 3, m.w);
}

// ---------------------------------------------------------------------------
// Kernel 3: C[M,N] = relu(A[M,K] @ W[K,N] + bias[N]) using fp32 WMMA 16x16x4.
// Grid: (ceil(M/16/8), N/64). Block: 256 threads = 8 waves.
//
// LDS holds W[:, n0:n0+64] *K-pair interleaved*:
//   lds_w[p*128 + n*2 + 0] = W[2p  ][n0+n]
//   lds_w[p*128 + n*2 + 1] = W[2p+1][n0+n]
// so a lane's 4x16 B fragment (two consecutive K rows at its column) is ONE
// contiguous 8-byte ds_load_b64 into an aligned VGPR pair -> no repacking
// moves between WMMAs (fixes the v_dual_mov churn seen in round 1).
//
// fp32 16x16x4 fragment layout (ISA 7.12.2):
//   A (16x4): lanes 0-15 -> M=lane, {K0,K1}; lanes 16-31 -> {K2,K3}
//   B (4x16): symmetric; lane n = lane&15, khalf selects K pair
//   C/D (16x16): lane n = lane&15, VGPR r -> M = r + 8*(lane>>4)
// ---------------------------------------------------------------------------
template <int K>
__global__ __launch_bounds__(256) void mlp_gemm_relu(
    const float* __restrict__ A, const float* __restrict__ W,
    const float* __restrict__ bias, float* __restrict__ C, int M, int N) {
    __shared__ float lds_w[K * 64];  // [K/2 pairs][64 cols][2]

    const int tid  = threadIdx.x;
    const int lane = tid & 31;
    const int wave = tid >> 5;
    const int n0   = blockIdx.y * 64;

    // Cooperative swizzled fill: iteration -> pair p, col c (2 cols/iter).
    // Reads two float2 (rows 2p, 2p+1), writes one interleaved float4.
    for (int idx = tid; idx < K * 16; idx += 256) {
        int p = idx >> 5;          // K-pair index (32 iters cover 64 cols)
        int c = (idx & 31) * 2;    // column within 64-slice
        float2 w0 = *(const float2*)&W[(size_t)(2 * p)     * N + n0 + c];
        float2 w1 = *(const float2*)&W[(size_t)(2 * p + 1) * N + n0 + c];
        float4 v  = make_float4(w0.x, w1.x, w0.y, w1.y);
        *(float4*)&lds_w[p * 128 + c * 2] = v;
    }
    __syncthreads();

    const int m0 = (blockIdx.x * 8 + wave) * 16;
    if (m0 >= M) return;  // wave-uniform: EXEC stays all-ones for WMMA

    const int mrow  = lane & 15;
    const int khalf = lane >> 4;  // 0: K pair {0,1}; 1: K pair {2,3}
    const float* Arow = A + (size_t)(m0 + mrow) * K;

    v8f acc[4] = {v8f{}, v8f{}, v8f{}, v8f{}};

#pragma unroll 4
    for (int kt = 0; kt < K / 4; ++kt) {
        const int kb = kt * 4 + khalf * 2;     // lane's K pair base
        const int p  = kt * 2 + khalf;         // pair index in LDS
        v2f a = *(const v2f*)&Arow[kb];        // A[m][kb], A[m][kb+1]
        const float* brow = &lds_w[p * 128 + (lane & 15) * 2];
#pragma unroll
        for (int nt = 0; nt < 4; ++nt) {
            v2f b = *(const v2f*)&brow[nt * 32];  // {W[kb][n], W[kb+1][n]}
            acc[nt] = __builtin_amdgcn_wmma_f32_16x16x4_f32(
                /*neg_a=*/false, a, /*neg_b=*/false, b,
                /*c_mod=*/(short)0, acc[nt],
                /*reuse_a=*/false, /*reuse_b=*/false);
        }
    }

    // bias + relu + store
#pragma unroll
    for (int nt = 0; nt < 4; ++nt) {
        int nc   = n0 + nt * 16 + (lane & 15);
        float bv = bias[nc];
#pragma unroll
        for (int r = 0; r < 8; ++r) {
            float v = acc[nt][r] + bv;
            v = v > 0.0f ? v : 0.0f;
            C[(size_t)(m0 + r + 8 * khalf) * N + nc] = v;
        }
    }
}

// ---------------------------------------------------------------------------
// Launch
// ---------------------------------------------------------------------------
extern "C" void kernel_launch(void* const* d_in, const int* in_sizes, int n_in,
                              void* d_out, int out_size, void* d_ws, size_t ws_size,
                              hipStream_t stream) {
    const float* node_feat  = (const float*)d_in[0];
    const int*   edge_index = (const int*)d_in[1];
    const float* edge_attr  = (const float*)d_in[2];
    const float* W1         = (const float*)d_in[3];
    const float* b1         = (const float*)d_in[4];
    const float* W2         = (const float*)d_in[5];
    const float* b2         = (const float*)d_in[6];

    const int N = in_sizes[0] / D_FEAT;       // 50000 nodes
    const int E = in_sizes[2] / D_FEAT;       // 800000 edges
    const int* src = edge_index;              // edge_index[0, :]
    const int* dst = edge_index + E;          // edge_index[1, :]

    float* h   = (float*)d_ws;                        // [N, 128]  25.6 MB
    float* mid = h + (size_t)N * D_FEAT;              // [N, 256]  51.2 MB
    float* out = (float*)d_out;                       // [N, 128]

    // 1) h = node_feat
    {
        int n4 = N * (D_FEAT / 4);
        init_h_kernel<<<(n4 + 255) / 256, 256, 0, stream>>>(node_feat, h, n4);
    }

    // 2) scatter-add edge messages
    {
        int blocks = (E + 7) / 8;  // 8 waves/block, one edge per wave
        edge_msg_kernel<<<blocks, 256, 0, stream>>>(node_feat, src, dst,
                                                    edge_attr, h, E);
    }

    // 3) mid = relu(h @ W1 + b1)   [N,128] x [128,256]
    {
        int mtiles = (N + 15) / 16;           // 3125
        dim3 grid((mtiles + 7) / 8, 256 / 64);
        mlp_gemm_relu<128><<<grid, 256, 0, stream>>>(h, W1, b1, mid, N, 256);
    }

    // 4) out = relu(mid @ W2 + b2) [N,256] x [256,128]  (outer ReLU fused)
    {
        int mtiles = (N + 15) / 16;
        dim3 grid((mtiles + 7) / 8, 128 / 64);
        mlp_gemm_relu<256><<<grid, 256, 0, stream>>>(mid, W2, b2, out, N, 128);
    }
}